// SAGELayer_12635793785118
// MI455X (gfx1250) — compile-verified
//
#include <hip/hip_runtime.h>

#define N_NODES 50000
#define N_EDGES 1600000
#define D 128
#define LDS_STRIDE 132  // 128 + 4 dword pad: row stride of 132 makes the 16-row
                        // strided A-fragment reads conflict-free on 64 banks

typedef __attribute__((ext_vector_type(2))) float v2f;
typedef __attribute__((ext_vector_type(8))) float v8f;

// ---------------------------------------------------------------------------
// Zero the workspace (agg + cnt) — must run every launch (ws is poisoned once
// and never re-poisoned; we re-zero so replays are deterministic).
// ---------------------------------------------------------------------------
__global__ __launch_bounds__(256) void zero_ws_kernel(float4* __restrict__ p, int n4) {
    int i = blockIdx.x * 256 + threadIdx.x;
    if (i < n4) p[i] = make_float4(0.f, 0.f, 0.f, 0.f);
}

// ---------------------------------------------------------------------------
// Phase 1: edge scatter. One wave per edge: lane L handles floats [4L, 4L+4).
// x[src] read is a fully coalesced 512B wave read; agg[dst] updates are 128
// contiguous global_atomic_add_f32 (L2-resident, L2 atomic units do the RMW).
// ---------------------------------------------------------------------------
__global__ __launch_bounds__(256) void scatter_accum_kernel(
    const float* __restrict__ x, const int* __restrict__ edge,
    float* __restrict__ agg, float* __restrict__ cnt) {
    unsigned gid  = blockIdx.x * 256u + threadIdx.x;
    unsigned e    = gid >> 5;   // edge id (one wave per edge)
    unsigned lane = gid & 31u;  // 4-float chunk within the 128-float row
    if (e >= N_EDGES) return;

    int src = edge[e];
    int dst = edge[N_EDGES + e];

    const float4 v = *(const float4*)(x + (size_t)src * D + lane * 4);
    float* a = agg + (size_t)dst * D + lane * 4;
    atomicAdd(a + 0, v.x);
    atomicAdd(a + 1, v.y);
    atomicAdd(a + 2, v.z);
    atomicAdd(a + 3, v.w);
    if (lane == 0) atomicAdd(cnt + dst, 1.0f);
}

// ---------------------------------------------------------------------------
// Phase 2: out = relu(mean @ Wl^T + bl + x @ Wr^T) via V_WMMA_F32_16X16X4_F32.
// Block = 16 rows x 128 cols; 8 waves, each wave owns a 16x16 output tile.
// A fragments from LDS (mean/x panels, bank-conflict-free via stride 132),
// B fragments (W[n][k]) straight from global (64KB, L2/L0 resident).
// fp32 WMMA keeps exact fp32 precision vs the fp32 reference.
// ---------------------------------------------------------------------------
__global__ __launch_bounds__(256) void sage_gemm_wmma_kernel(
    const float* __restrict__ x,   const float* __restrict__ agg,
    const float* __restrict__ cnt, const float* __restrict__ Wl,
    const float* __restrict__ bl,  const float* __restrict__ Wr,
    float* __restrict__ out) {
    __shared__ float sMean[16 * LDS_STRIDE];
    __shared__ float sX   [16 * LDS_STRIDE];

    const int tid     = threadIdx.x;
    const int rowBase = blockIdx.x * 16;   // 50000 = 3125 * 16, exact

    // ---- stage 16x128 panels of mean (= agg * 1/max(cnt,1)) and x into LDS
    {
        const int r = tid >> 4;             // 0..15 (row in tile)
        const int c = (tid & 15) * 8;       // 0..120 (8 floats per thread)
        const int g = rowBase + r;
        const float inv = 1.0f / fmaxf(cnt[g], 1.0f);
        const float4* ap = (const float4*)(agg + (size_t)g * D + c);
        const float4* xp = (const float4*)(x   + (size_t)g * D + c);
        float4 a0 = ap[0], a1 = ap[1];
        float4 x0 = xp[0], x1 = xp[1];
        float* sm = sMean + r * LDS_STRIDE + c;
        float* sx = sX    + r * LDS_STRIDE + c;
        sm[0] = a0.x * inv; sm[1] = a0.y * inv; sm[2] = a0.z * inv; sm[3] = a0.w * inv;
        sm[4] = a1.x * inv; sm[5] = a1.y * inv; sm[6] = a1.z * inv; sm[7] = a1.w * inv;
        sx[0] = x0.x; sx[1] = x0.y; sx[2] = x0.z; sx[3] = x0.w;
        sx[4] = x1.x; sx[5] = x1.y; sx[6] = x1.z; sx[7] = x1.w;
    }
    __syncthreads();

    // ---- wave-level WMMA: wave w computes cols [16w, 16w+16) of the 16 rows
    const int lane = tid & 31;
    const int wave = tid >> 5;
    const int col0 = wave * 16;
    const int n    = lane & 15;        // N index (A row index uses same lane&15)
    const int kh   = (lane >> 4) * 2;  // K sub-offset for this lane half

    // C/D layout: VGPR v, lanes 0-15 -> M=v, lanes 16-31 -> M=v+8; N = lane&15.
    v8f acc;
    const float bv = bl[col0 + n];
#pragma unroll
    for (int v = 0; v < 8; ++v) acc[v] = bv;

    const float* wlp = Wl + (size_t)(col0 + n) * D + kh;  // B[k][n] = W[n][k]
    const float* wrp = Wr + (size_t)(col0 + n) * D + kh;
    const float* amp = sMean + n * LDS_STRIDE + kh;       // A row m = lane&15
    const float* axp = sX    + n * LDS_STRIDE + kh;

#pragma unroll 8
    for (int k = 0; k < D; k += 4) {
        v2f a = *(const v2f*)(amp + k);
        v2f b = *(const v2f*)(wlp + k);
        acc = __builtin_amdgcn_wmma_f32_16x16x4_f32(
            false, a, false, b, (short)0, acc, false, false);
    }
#pragma unroll 8
    for (int k = 0; k < D; k += 4) {
        v2f a = *(const v2f*)(axp + k);
        v2f b = *(const v2f*)(wrp + k);
        acc = __builtin_amdgcn_wmma_f32_16x16x4_f32(
            false, a, false, b, (short)0, acc, false, false);
    }

    // ---- fused ReLU + store
    float* op = out + (size_t)rowBase * D + col0 + n;
    const int mBase = (lane >> 4) * 8;
#pragma unroll
    for (int v = 0; v < 8; ++v) {
        op[(size_t)(mBase + v) * D] = fmaxf(acc[v], 0.0f);
    }
}

// ---------------------------------------------------------------------------
extern "C" void kernel_launch(void* const* d_in, const int* in_sizes, int n_in,
                              void* d_out, int out_size, void* d_ws, size_t ws_size,
                              hipStream_t stream) {
    (void)in_sizes; (void)n_in; (void)out_size; (void)ws_size;
    const float* x    = (const float*)d_in[0];
    const int*   edge = (const int*)d_in[1];   // [2, N_EDGES] int32
    const float* Wl   = (const float*)d_in[2];
    const float* bl   = (const float*)d_in[3];
    const float* Wr   = (const float*)d_in[4];
    float*       out  = (float*)d_out;

    float* agg = (float*)d_ws;                       // [N_NODES, D] = 25.6 MB
    float* cnt = agg + (size_t)N_NODES * D;          // [N_NODES]    = 200 KB

    // zero agg + cnt (6,450,000 floats = 1,612,500 float4)
    const int n4 = (N_NODES * D + N_NODES) / 4;
    zero_ws_kernel<<<(n4 + 255) / 256, 256, 0, stream>>>((float4*)d_ws, n4);

    // scatter: one wave per edge -> 1.6M * 32 threads
    const int scatterBlocks = (N_EDGES * 32) / 256;  // 200,000
    scatter_accum_kernel<<<scatterBlocks, 256, 0, stream>>>(x, edge, agg, cnt);

    // fused dual-GEMM + bias + relu: 3125 blocks x (16 rows, 8 wmma waves)
    sage_gemm_wmma_kernel<<<N_NODES / 16, 256, 0, stream>>>(
        x, agg, cnt, Wl, bl, Wr, out);
}